// MultiHeadAttention_63660005261376
// MI455X (gfx1250) — compile-verified
//
#include <hip/hip_runtime.h>

typedef float v2f __attribute__((ext_vector_type(2)));
typedef float v8f __attribute__((ext_vector_type(8)));

constexpr int B = 2, S = 2048, D = 768, H = 12, HD = 64;
constexpr int BH = B * H;
constexpr size_t QSZ = (size_t)B * H * S * HD;   // elements per Q/K/V buffer
constexpr float SCALE = 0.125f;                  // 1/sqrt(HD=64)

static __device__ __forceinline__ v8f wmma4(v2f a, v2f b, v8f c) {
  // D = A(16x4,f32) * B(4x16,f32) + C(16x16,f32)  -> v_wmma_f32_16x16x4_f32
  return __builtin_amdgcn_wmma_f32_16x16x4_f32(false, a, false, b, (short)0, c,
                                               false, false);
}

// ---------------------------------------------------------------------------
// Kernel 1: Q/K/V projection.  grid=(S/64, 3*B*H), block=128 (4 waves).
// Each wave owns one 16-row strip (s_tile = blockIdx.x*4 + wave) and computes
// the full 16x64 output: A fragment loaded once per k-step, reused for all
// 4 n-tiles; B fragments (W) are identical across the block's 4 waves.
// ---------------------------------------------------------------------------
__global__ void __launch_bounds__(128)
qkv_proj_kernel(const float* __restrict__ X, const float* __restrict__ wq,
                const float* __restrict__ wk, const float* __restrict__ wv,
                float* __restrict__ ws) {
  const int lane   = threadIdx.x & 31;
  const int wave   = threadIdx.x >> 5;
  const int lane15 = lane & 15;
  const int koff   = (lane >= 16) ? 2 : 0;
  const int hi8    = (lane >= 16) ? 8 : 0;

  const int s0  = (blockIdx.x * 4 + wave) * 16;
  const int bid = blockIdx.y;                // 0..3*BH-1
  const int qkv = bid / BH;
  const int bh  = bid % BH;
  const int b   = bh / H, h = bh % H;

  const float* W = (qkv == 0) ? wq : (qkv == 1) ? wk : wv;  // [H, D, HD]
  float* outp = ws + (size_t)qkv * QSZ;                     // [B,H,S,HD]

  const size_t xrow  = (size_t)(b * S + s0 + lane15) * D;
  const size_t wbase = (size_t)h * D * HD;

  v8f acc[4] = {};
  for (int ks = 0; ks < D / 4; ++ks) {
    const int k = ks * 4 + koff;
    v2f a; a.x = X[xrow + k]; a.y = X[xrow + k + 1];
#pragma unroll
    for (int g = 0; g < 4; ++g) {
      const int n = g * 16 + lane15;
      v2f bb; bb.x = W[wbase + (size_t)k * HD + n];
              bb.y = W[wbase + (size_t)(k + 1) * HD + n];
      acc[g] = wmma4(a, bb, acc[g]);
    }
  }
#pragma unroll
  for (int g = 0; g < 4; ++g)
#pragma unroll
    for (int v = 0; v < 8; ++v) {
      const int r = v + hi8;
      outp[(size_t)(bh * S + s0 + r) * HD + g * 16 + lane15] = acc[g][v];
    }
}

// ---------------------------------------------------------------------------
// Kernel 2: scores = Q K^T * scale, mask, softmax -> normalized weights.
// grid=(S/16, B*H), block=512 (16 waves).  Block owns a 16-row x 2048-col
// strip; wave w keeps 8 score tiles (n-tiles w, w+16, ..., w+112) in VGPRs.
// Row max / sum-of-exp reduced across waves through LDS.
// ---------------------------------------------------------------------------
__global__ void __launch_bounds__(512)
attn_softmax_kernel(const float* __restrict__ Q, const float* __restrict__ Km,
                    const int* __restrict__ mask, float* __restrict__ wout) {
  __shared__ float red[16][256];
  __shared__ float rowstat[16];

  const int tid    = threadIdx.x;
  const int lane   = tid & 31;
  const int wave   = tid >> 5;               // 0..15
  const int lane15 = lane & 15;
  const int koff   = (lane >= 16) ? 2 : 0;
  const int hi8    = (lane >= 16) ? 8 : 0;
  const float NEG_INF = -__builtin_inff();

  const int s0 = blockIdx.x * 16;
  const int bh = blockIdx.y;

  const size_t qbase = (size_t)bh * S * HD;
  const size_t qrow  = qbase + (size_t)(s0 + lane15) * HD;

  // Preload the 16 A-fragments covering K=64 (shared by all 8 tiles).
  v2f aA[16];
#pragma unroll
  for (int ks = 0; ks < 16; ++ks) {
    const int k = ks * 4 + koff;
    aA[ks].x = Q[qrow + k];
    aA[ks].y = Q[qrow + k + 1];
  }

  v8f acc[8];
#pragma unroll
  for (int t = 0; t < 8; ++t) {
    const int    nt   = wave + 16 * t;
    const size_t krow = qbase + (size_t)(nt * 16 + lane15) * HD;  // B = K^T
    v8f c = {};
#pragma unroll
    for (int ks = 0; ks < 16; ++ks) {
      const int k = ks * 4 + koff;
      v2f bb; bb.x = Km[krow + k]; bb.y = Km[krow + k + 1];
      c = wmma4(aA[ks], bb, c);
    }
    acc[t] = c;
  }

  // --- scale + mask + per-lane running max ---
  float pmax[8];
#pragma unroll
  for (int v = 0; v < 8; ++v) pmax[v] = NEG_INF;
#pragma unroll
  for (int t = 0; t < 8; ++t) {
    const int col = (wave + 16 * t) * 16 + lane15;
#pragma unroll
    for (int v = 0; v < 8; ++v) {
      const int r = v + hi8;
      const int m = mask[(size_t)(s0 + r) * S + col];
      const float val = m ? acc[t][v] * SCALE : NEG_INF;
      acc[t][v] = val;
      pmax[v]   = fmaxf(pmax[v], val);
    }
  }
#pragma unroll
  for (int v = 0; v < 8; ++v) red[v + hi8][wave * 16 + lane15] = pmax[v];
  __syncthreads();
  if (tid < 16) {
    float m = NEG_INF;
    for (int j = 0; j < 256; ++j) m = fmaxf(m, red[tid][j]);
    rowstat[tid] = m;
  }
  __syncthreads();
  float rmax[8];
#pragma unroll
  for (int v = 0; v < 8; ++v) rmax[v] = rowstat[v + hi8];

  // --- exp + per-lane partial sums ---
  float psum[8];
#pragma unroll
  for (int v = 0; v < 8; ++v) psum[v] = 0.0f;
#pragma unroll
  for (int t = 0; t < 8; ++t) {
#pragma unroll
    for (int v = 0; v < 8; ++v) {
      const float e = __expf(acc[t][v] - rmax[v]);
      acc[t][v] = e;
      psum[v] += e;
    }
  }
#pragma unroll
  for (int v = 0; v < 8; ++v) red[v + hi8][wave * 16 + lane15] = psum[v];
  __syncthreads();
  if (tid < 16) {
    float ssum = 0.0f;
    for (int j = 0; j < 256; ++j) ssum += red[tid][j];
    rowstat[tid] = ssum;
  }
  __syncthreads();

  // --- normalize and write weights once ---
#pragma unroll
  for (int v = 0; v < 8; ++v) {
    const int    r    = v + hi8;
    const float  inv  = 1.0f / rowstat[r];
    const size_t orow = ((size_t)bh * S + s0 + r) * S;
#pragma unroll
    for (int t = 0; t < 8; ++t) {
      const int col = (wave + 16 * t) * 16 + lane15;
      wout[orow + col] = acc[t][v] * inv;
    }
  }
}

// ---------------------------------------------------------------------------
// Kernel 3: ctx = weights @ V, written in head-concat layout [B,S,D].
// grid=(S/64, B*H), block=128 (4 waves).  Each wave owns a 16-row strip and
// computes all 4 HD n-tiles.  The 402MB weights stream is read exactly once
// (rows wave-exclusive) -> non-temporal loads keep it out of L2's way, and
// global_prefetch_b8 runs 4KB ahead of the stream.  V fragments are shared
// by the block's 4 waves.
// ---------------------------------------------------------------------------
__global__ void __launch_bounds__(128)
attn_ctx_kernel(const float* __restrict__ Wt, const float* __restrict__ V,
                float* __restrict__ concat) {
  const int lane   = threadIdx.x & 31;
  const int wave   = threadIdx.x >> 5;
  const int lane15 = lane & 15;
  const int koff   = (lane >= 16) ? 2 : 0;
  const int hi8    = (lane >= 16) ? 8 : 0;

  const int s0 = (blockIdx.x * 4 + wave) * 16;
  const int bh = blockIdx.y;
  const int b  = bh / H, h = bh % H;

  const size_t wrow  = ((size_t)bh * S + s0 + lane15) * S;
  const size_t vbase = (size_t)bh * S * HD;

  v8f acc[4] = {};
  for (int ks = 0; ks < S / 4; ++ks) {
    const int k = ks * 4 + koff;
    if ((ks & 15) == 0)
      __builtin_prefetch(Wt + wrow + k + 1024, 0, 1);  // 4KB ahead of stream
    v2f a;
    a.x = __builtin_nontemporal_load(Wt + wrow + k);
    a.y = __builtin_nontemporal_load(Wt + wrow + k + 1);
#pragma unroll
    for (int g = 0; g < 4; ++g) {
      const int n = g * 16 + lane15;
      v2f bb; bb.x = V[vbase + (size_t)k * HD + n];
              bb.y = V[vbase + (size_t)(k + 1) * HD + n];
      acc[g] = wmma4(a, bb, acc[g]);
    }
  }
#pragma unroll
  for (int g = 0; g < 4; ++g)
#pragma unroll
    for (int v = 0; v < 8; ++v) {
      const int r = v + hi8;
      concat[(size_t)(b * S + s0 + r) * D + h * HD + g * 16 + lane15] =
          acc[g][v];
    }
}

// ---------------------------------------------------------------------------
// Kernel 4: out = concat @ wo.  grid=(S/64, 12, B), block=128 (4 waves).
// Wave owns a 16-row strip; computes 4 consecutive n-tiles (blockIdx.y picks
// which 64-column group of D=768).  A fragment (concat row) loaded once per
// k-step, reused 4x; wo fragments shared across the block's waves.
// ---------------------------------------------------------------------------
__global__ void __launch_bounds__(128)
out_proj_kernel(const float* __restrict__ Cc, const float* __restrict__ wo,
                float* __restrict__ out) {
  const int lane   = threadIdx.x & 31;
  const int wave   = threadIdx.x >> 5;
  const int lane15 = lane & 15;
  const int koff   = (lane >= 16) ? 2 : 0;
  const int hi8    = (lane >= 16) ? 8 : 0;

  const int s0 = (blockIdx.x * 4 + wave) * 16;
  const int n0 = blockIdx.y * 64;            // 64-col group of D
  const int b  = blockIdx.z;

  const size_t crow = (size_t)(b * S + s0 + lane15) * D;

  v8f acc[4] = {};
  for (int ks = 0; ks < D / 4; ++ks) {
    const int k = ks * 4 + koff;
    v2f a; a.x = Cc[crow + k]; a.y = Cc[crow + k + 1];
#pragma unroll
    for (int g = 0; g < 4; ++g) {
      const int n = n0 + g * 16 + lane15;
      v2f bb; bb.x = wo[(size_t)k * D + n];
              bb.y = wo[(size_t)(k + 1) * D + n];
      acc[g] = wmma4(a, bb, acc[g]);
    }
  }
#pragma unroll
  for (int g = 0; g < 4; ++g)
#pragma unroll
    for (int v = 0; v < 8; ++v) {
      const int r = v + hi8;
      out[(size_t)(b * S + s0 + r) * D + n0 + g * 16 + lane15] = acc[g][v];
    }
}

// ---------------------------------------------------------------------------
extern "C" void kernel_launch(void* const* d_in, const int* in_sizes, int n_in,
                              void* d_out, int out_size, void* d_ws,
                              size_t ws_size, hipStream_t stream) {
  (void)in_sizes; (void)n_in; (void)out_size; (void)ws_size;

  const float* X    = (const float*)d_in[0];  // [B,S,D]
  const float* wq   = (const float*)d_in[1];  // [H,D,HD]
  const float* wk   = (const float*)d_in[2];
  const float* wv   = (const float*)d_in[3];
  const float* wo   = (const float*)d_in[4];  // [D,D]
  const int*   mask = (const int*)d_in[5];    // [1,S,S]

  float* out     = (float*)d_out;                    // [B,S,D]
  float* weights = out + (size_t)B * S * D;          // [B,H,S,S]

  float* ws = (float*)d_ws;
  float* Q  = ws;                 // [B,H,S,HD]
  float* K  = ws + QSZ;
  float* V  = ws + 2 * QSZ;
  float* Cc = ws + 3 * QSZ;       // concat [B,S,D]

  qkv_proj_kernel    <<<dim3(S / 64, 3 * BH), 128, 0, stream>>>(X, wq, wk, wv, ws);
  attn_softmax_kernel<<<dim3(S / 16, BH),     512, 0, stream>>>(Q, K, mask, weights);
  attn_ctx_kernel    <<<dim3(S / 64, BH),     128, 0, stream>>>(weights, V, Cc);
  out_proj_kernel    <<<dim3(S / 64, 12, B),  128, 0, stream>>>(Cc, wo, out);
}